// VanillaNeedleRIN_75892072121022
// MI455X (gfx1250) — compile-verified
//
#include <hip/hip_runtime.h>
#include <hip/hip_bf16.h>

typedef __attribute__((ext_vector_type(2))) float v2f;
typedef __attribute__((ext_vector_type(8))) float v8f;

#define PHI_F     1.6180339887498949f
#define TWO_PI_F  6.28318530717958648f
#define LUT_N     4096
#define LUT_SCALE 651.8986469044033f       /* LUT_N / 2pi  */
#define LUT_STEP  0.0015339807878856412f   /* 2pi / LUT_N  */

#define BB   64      /* batch */
#define SS   4096    /* sequence */
#define DD   64      /* half embedding dim */
#define NN   128     /* hidden */
#define NSIG 32000   /* output signals */

// ---------------------------------------------------------------------------
// LUT lookup with linear interpolation (matches reference _lookup_sin_cos)
// ---------------------------------------------------------------------------
__device__ __forceinline__ void lut_lookup(const float* __restrict__ s_sin,
                                           const float* __restrict__ s_cos,
                                           float th, float& s, float& c) {
    float pos  = th * LUT_SCALE;
    float i0   = floorf(pos);
    float frac = pos - i0;
    int idx0 = ((int)i0) & (LUT_N - 1);
    int idx1 = (idx0 + 1) & (LUT_N - 1);
    float omf = 1.0f - frac;
    s = s_sin[idx0] * omf + s_sin[idx1] * frac;
    c = s_cos[idx0] * omf + s_cos[idx1] * frac;
}

// ---------------------------------------------------------------------------
// Stage 1: the time recurrence. One thread per (b,d) chain, LUT in LDS.
// grid = 32 blocks x 128 threads  (B*D = 4096 chains)
// ---------------------------------------------------------------------------
__global__ __launch_bounds__(128) void rin_recurrence(
    const int*   __restrict__ ids,
    const float* __restrict__ emb,
    float* __restrict__ hr_out,
    float* __restrict__ hi_out) {
    __shared__ float s_sin[LUT_N];
    __shared__ float s_cos[LUT_N];
    __shared__ float s_tphi[SS];

    int tid = threadIdx.x;
    for (int i = tid; i < LUT_N; i += 128) {
        float ang = (float)i * LUT_STEP;
        s_sin[i]  = sinf(ang);
        s_cos[i]  = cosf(ang);
        s_tphi[i] = fmodf((float)i * PHI_F, TWO_PI_F);
    }
    __syncthreads();

    int g = blockIdx.x * 128 + tid;
    int b = g >> 6;
    int d = g & 63;
    const int* idrow = ids + b * SS;

    float hr = 0.0f, hi = 0.0f;
    int   id = idrow[0];
    float w  = emb[id * (2 * DD) + d];
    float bt = emb[id * (2 * DD) + DD + d];

    for (int t = 0; t < SS; ++t) {
        // prefetch next timestep's embedding (independent of the h chain)
        float wn = 0.0f, bn = 0.0f;
        if (t + 1 < SS) {
            int idn = idrow[t + 1];
            wn = emb[idn * (2 * DD) + d];
            bn = emb[idn * (2 * DD) + DD + d];
        }
        float wl   = 1.0f + fabsf(w);
        float inv  = 1.0f / wl;
        float tphi = s_tphi[t];
        float thr  = hr * inv + bt + tphi;
        float thi  = hi * inv + bt + tphi;
        float sr, cr, si, ci;
        lut_lookup(s_sin, s_cos, thr, sr, cr);
        lut_lookup(s_sin, s_cos, thi, si, ci);
        hr = cr * ci - sr * si;
        hi = cr * si + sr * ci;
        w = wn; bt = bn;
    }
    hr_out[g] = hr;
    hi_out[g] = hi;
}

// ---------------------------------------------------------------------------
// Stage 2a: layer input projection + rotation + swish -> (vr, vi) (B x N)
// grid = 32 blocks x 256 threads (8192 elements)
// ---------------------------------------------------------------------------
__global__ __launch_bounds__(256) void rin_layer_in(
    const float* __restrict__ xr, const float* __restrict__ xi,
    const float* __restrict__ winr, const float* __restrict__ wini,
    const float* __restrict__ lw,   const float* __restrict__ lb,
    float* __restrict__ vr, float* __restrict__ vi) {
    int e = blockIdx.x * 256 + threadIdx.x;      // 0 .. B*N-1
    int brow = e >> 7;
    int n    = e & (NN - 1);

    const float* xr_row = xr + brow * DD;
    const float* xi_row = xi + brow * DD;
    float ur = 0.0f, ui = 0.0f;
#pragma unroll 8
    for (int k = 0; k < DD; ++k) {
        float wr = winr[k * NN + n];
        float wi = wini[k * NN + n];
        float ar = xr_row[k];
        float ai = xi_row[k];
        ur += ar * wr - ai * wi;
        ui += ar * wi + ai * wr;
    }

    // theta depends only on n: wrap(t_last)/wavelength + b
    float t_last = (float)((double)(SS - 1) * 1.618033988749895); // f32((S-1)*PHI)
    float tphase = fmodf(t_last, TWO_PI_F);
    float wlen   = 1.0f + fabsf(lw[n]);
    float theta  = tphase / wlen + lb[n];

    float pos  = theta * LUT_SCALE;
    float i0   = floorf(pos);
    float frac = pos - i0;
    int idx0 = ((int)i0) & (LUT_N - 1);
    int idx1 = (idx0 + 1) & (LUT_N - 1);
    float omf = 1.0f - frac;
    float s = sinf((float)idx0 * LUT_STEP) * omf + sinf((float)idx1 * LUT_STEP) * frac;
    float c = cosf((float)idx0 * LUT_STEP) * omf + cosf((float)idx1 * LUT_STEP) * frac;

    float pr = ur * c - ui * s;
    float pi = ur * s + ui * c;
    // swish
    vr[e] = pr * (1.0f / (1.0f + expf(-pr)));
    vi[e] = pi * (1.0f / (1.0f + expf(-pi)));
}

// ---------------------------------------------------------------------------
// Stage 2b: layer output projection -> new (xr, xi) (B x D)
// grid = 16 blocks x 256 threads (4096 elements)
// ---------------------------------------------------------------------------
__global__ __launch_bounds__(256) void rin_layer_out(
    const float* __restrict__ vr, const float* __restrict__ vi,
    const float* __restrict__ woutr, const float* __restrict__ wouti,
    float* __restrict__ xr, float* __restrict__ xi) {
    int e = blockIdx.x * 256 + threadIdx.x;      // 0 .. B*D-1
    int brow = e >> 6;
    int d    = e & (DD - 1);

    const float* vr_row = vr + brow * NN;
    const float* vi_row = vi + brow * NN;
    float outr = 0.0f, outi = 0.0f;
#pragma unroll 8
    for (int n = 0; n < NN; ++n) {
        float wr = woutr[n * DD + d];
        float wi = wouti[n * DD + d];
        float ar = vr_row[n];
        float ai = vi_row[n];
        outr += ar * wr - ai * wi;
        outi += ar * wi + ai * wr;
    }
    xr[e] = outr;
    xi[e] = outi;
}

// ---------------------------------------------------------------------------
// Stage 3: out = [xr|xi](64x128) @ W(128x32000) via V_WMMA_F32_16X16X4_F32
//   W[k][n]  = out_wr[k][n]+out_wi[k][n]          (k < 64)
//            = out_wr[k-64][n]-out_wi[k-64][n]    (k >= 64)
// One wave per 16-wide N tile, 4 M-tile accumulators -> W streamed once.
// grid = 250 blocks x 256 threads (2000 N tiles)
// ---------------------------------------------------------------------------
__global__ __launch_bounds__(256) void rin_out_gemm(
    const float* __restrict__ xr, const float* __restrict__ xi,
    const float* __restrict__ owr, const float* __restrict__ owi,
    float* __restrict__ out) {
    __shared__ float Xs[BB * 2 * DD];            // 64 x 128, row-major, 32 KB
    for (int i = threadIdx.x; i < BB * 2 * DD; i += 256) {
        int m = i >> 7;
        int k = i & 127;
        Xs[i] = (k < DD) ? xr[m * DD + k] : xi[m * DD + (k - DD)];
    }
    __syncthreads();

    int lane  = threadIdx.x & 31;
    int wave  = threadIdx.x >> 5;
    int ntile = blockIdx.x * 8 + wave;
    int ncol  = ntile * 16 + (lane & 15);
    int khalf = lane >> 4;                        // 0: K pair {0,1}, 1: K pair {2,3}
    int mrow  = lane & 15;

    v8f acc0 = {}, acc1 = {}, acc2 = {}, acc3 = {};

    for (int k0 = 0; k0 < 2 * DD; k0 += 4) {
        int kb = k0 + 2 * khalf;                  // kb, kb+1 never straddle 64
        v2f bfrag;
        if (kb < DD) {
            bfrag[0] = owr[kb * NSIG + ncol]       + owi[kb * NSIG + ncol];
            bfrag[1] = owr[(kb + 1) * NSIG + ncol] + owi[(kb + 1) * NSIG + ncol];
        } else {
            int kk = kb - DD;
            bfrag[0] = owr[kk * NSIG + ncol]       - owi[kk * NSIG + ncol];
            bfrag[1] = owr[(kk + 1) * NSIG + ncol] - owi[(kk + 1) * NSIG + ncol];
        }

        v2f a0, a1, a2, a3;
        a0[0] = Xs[(0 * 16 + mrow) * 128 + kb];  a0[1] = Xs[(0 * 16 + mrow) * 128 + kb + 1];
        a1[0] = Xs[(1 * 16 + mrow) * 128 + kb];  a1[1] = Xs[(1 * 16 + mrow) * 128 + kb + 1];
        a2[0] = Xs[(2 * 16 + mrow) * 128 + kb];  a2[1] = Xs[(2 * 16 + mrow) * 128 + kb + 1];
        a3[0] = Xs[(3 * 16 + mrow) * 128 + kb];  a3[1] = Xs[(3 * 16 + mrow) * 128 + kb + 1];

        acc0 = __builtin_amdgcn_wmma_f32_16x16x4_f32(false, a0, false, bfrag, (short)0, acc0, false, false);
        acc1 = __builtin_amdgcn_wmma_f32_16x16x4_f32(false, a1, false, bfrag, (short)0, acc1, false, false);
        acc2 = __builtin_amdgcn_wmma_f32_16x16x4_f32(false, a2, false, bfrag, (short)0, acc2, false, false);
        acc3 = __builtin_amdgcn_wmma_f32_16x16x4_f32(false, a3, false, bfrag, (short)0, acc3, false, false);
    }

    // C/D layout: VGPR i -> M = i + 8*khalf, N = lane&15
#pragma unroll
    for (int i = 0; i < 8; ++i) {
        int r = 8 * khalf + i;
        out[(0 * 16 + r) * NSIG + ncol] = acc0[i];
        out[(1 * 16 + r) * NSIG + ncol] = acc1[i];
        out[(2 * 16 + r) * NSIG + ncol] = acc2[i];
        out[(3 * 16 + r) * NSIG + ncol] = acc3[i];
    }
}

// ---------------------------------------------------------------------------
extern "C" void kernel_launch(void* const* d_in, const int* in_sizes, int n_in,
                              void* d_out, int out_size, void* d_ws, size_t ws_size,
                              hipStream_t stream) {
    const int*   ids    = (const int*)  d_in[0];
    const float* emb    = (const float*)d_in[1];
    const float* win_r  = (const float*)d_in[2];
    const float* win_i  = (const float*)d_in[3];
    const float* wout_r = (const float*)d_in[4];
    const float* wout_i = (const float*)d_in[5];
    const float* lw     = (const float*)d_in[6];
    const float* lb     = (const float*)d_in[7];
    const float* out_wr = (const float*)d_in[8];
    const float* out_wi = (const float*)d_in[9];
    float* out = (float*)d_out;

    float* ws = (float*)d_ws;
    float* xr = ws;                    // B*D   = 4096
    float* xi = ws + 4096;             // B*D   = 4096
    float* vr = ws + 8192;             // B*N   = 8192
    float* vi = ws + 16384;            // B*N   = 8192

    rin_recurrence<<<32, 128, 0, stream>>>(ids, emb, xr, xi);

    for (int l = 0; l < 2; ++l) {
        rin_layer_in<<<32, 256, 0, stream>>>(
            xr, xi,
            win_r + (size_t)l * DD * NN, win_i + (size_t)l * DD * NN,
            lw + (size_t)l * NN, lb + (size_t)l * NN,
            vr, vi);
        rin_layer_out<<<16, 256, 0, stream>>>(
            vr, vi,
            wout_r + (size_t)l * NN * DD, wout_i + (size_t)l * NN * DD,
            xr, xi);
    }

    rin_out_gemm<<<250, 256, 0, stream>>>(xr, xi, out_wr, out_wi, out);
}